// UniModalLSTMModel_8005819039927
// MI455X (gfx1250) — compile-verified
//
#include <hip/hip_runtime.h>
#include <hip/hip_bf16.h>

// CDNA5 / gfx1250 fused 2-layer LSTM + FC using V_WMMA_F32_16X16X4_F32.
// One wave32 owns 16 batch rows; whole recurrence lives in registers.
// Software-pipelined: x(t+1)@W1 (4 WMMAs) and h2@U2 are computed off the
// serial h-recurrence path; each layer's critical segment is ONE WMMA +
// gates + a per-wave LDS transpose bounce.
//
// WMMA f32 16x16x4 register layouts used:
//   A (16x4):  lane m (0-15) holds {A[m][0],A[m][1]}, lane m+16 holds {A[m][2],A[m][3]}
//   B (4x16):  VGPR0: lanes0-15 = B[0][n], lanes16-31 = B[2][n]; VGPR1: B[1][n]/B[3][n]
//   C/D (16x16): VGPR r: lanes0-15 = D[r][n], lanes16-31 = D[r+8][n]
// Gate columns permuted to unit-major (col n -> hidden u = n>>2, gate g = n&3) so the
// i/f/g/o combine is 3 DPP quad_perm broadcasts (canonical result in lane 0 of each quad).

typedef __attribute__((ext_vector_type(2))) float v2f;
typedef __attribute__((ext_vector_type(8))) float v8f;

#define B_SZ 4096
#define T_SZ 512
#define F_SZ 16
#define G_SZ 16   // 4*H, H = 4
#define WAVES_PER_BLOCK 4

__device__ __forceinline__ float fast_sigmoid(float x) {
    return __builtin_amdgcn_rcpf(1.0f + __expf(-x));   // mul+exp+add+rcp
}

template <int CTRL>
__device__ __forceinline__ float quad_bcast(float v) {
    int i = __builtin_bit_cast(int, v);
    i = __builtin_amdgcn_update_dpp(0, i, CTRL, 0xf, 0xf, true);
    return __builtin_bit_cast(float, i);
}

__global__ __launch_bounds__(32 * WAVES_PER_BLOCK)
void lstm2_fc_wmma_f32(const float* __restrict__ x,     // [B, T, F]
                       const float* __restrict__ W1,    // [16, 16]
                       const float* __restrict__ U1,    // [4, 16]
                       const float* __restrict__ b1,    // [16]
                       const float* __restrict__ W2,    // [4, 16]
                       const float* __restrict__ U2,    // [4, 16]
                       const float* __restrict__ b2,    // [16]
                       const float* __restrict__ Wfc,   // [4]
                       const float* __restrict__ bfc,   // [1]
                       float* __restrict__ out)         // [B]
{
    __shared__ float lds1[WAVES_PER_BLOCK][64];  // h1 transpose scratch (per wave)
    __shared__ float lds2[WAVES_PER_BLOCK][64];  // h2 transpose scratch (per wave)

    const int lane   = threadIdx.x & 31;
    const int wv     = threadIdx.x >> 5;
    const int wave   = blockIdx.x * WAVES_PER_BLOCK + wv;
    const int b0     = wave * 16;
    const int n      = lane & 15;                // column index (B/C/D layouts)
    const int m      = lane & 15;                // row index (A layout)
    const int hi     = lane >> 4;
    const int u      = n >> 2;                   // hidden unit (unit-major columns)
    const int g      = n & 3;                    // gate: 0=i,1=f,2=g,3=o
    const int ocol   = g * 4 + u;                // original Keras [i,f,g,o] column

    // ---- weights in B layout, gate columns permuted to unit-major ----
    v2f w1b[4];
    #pragma unroll
    for (int j = 0; j < 4; ++j) {
        const int k0 = 4 * j + 2 * hi;
        w1b[j].x = W1[(k0 + 0) * G_SZ + ocol];
        w1b[j].y = W1[(k0 + 1) * G_SZ + ocol];
    }
    v2f u1b, w2b, u2b;
    {
        const int k0 = 2 * hi;
        u1b.x = U1[(k0 + 0) * G_SZ + ocol];  u1b.y = U1[(k0 + 1) * G_SZ + ocol];
        w2b.x = W2[(k0 + 0) * G_SZ + ocol];  w2b.y = W2[(k0 + 1) * G_SZ + ocol];
        u2b.x = U2[(k0 + 0) * G_SZ + ocol];  u2b.y = U2[(k0 + 1) * G_SZ + ocol];
    }
    v8f cbias1, cbias2;
    {
        const float bb1 = b1[ocol], bb2 = b2[ocol];
        #pragma unroll
        for (int r = 0; r < 8; ++r) { cbias1[r] = bb1; cbias2[r] = bb2; }
    }
    const float act_sel = (g == 2) ? 1.0f : 0.0f;   // relu for gate "g", else sigmoid

    // ---- recurrent state ----
    float c1[8], c2[8];
    #pragma unroll
    for (int r = 0; r < 8; ++r) { c1[r] = 0.f; c2[r] = 0.f; }
    v2f h1a = {0.f, 0.f}, h2a = {0.f, 0.f};      // hidden state in WMMA A layout

    const float* rowbase = x + (size_t)(b0 + m) * (T_SZ * F_SZ);
    float hbuf[8];

    // ---- prologue: zx = b1 + x(0)@W1 ; zh2 = b2 + h2(0)@U2 ----
    v2f xa[4];
    #pragma unroll
    for (int j = 0; j < 4; ++j)
        xa[j] = *(const v2f*)(rowbase + 4 * j + 2 * hi);
    v8f zx = cbias1;
    #pragma unroll
    for (int j = 0; j < 4; ++j)
        zx = __builtin_amdgcn_wmma_f32_16x16x4_f32(false, xa[j], false, w1b[j],
                                                   (short)0, zx, false, false);
    v8f zh2 = __builtin_amdgcn_wmma_f32_16x16x4_f32(false, h2a, false, u2b,
                                                    (short)0, cbias2, false, false);

    for (int t = 0; t < T_SZ; ++t) {
        // ======== layer 1 critical segment: z1 = zx + h1@U1 ========
        v8f z1 = __builtin_amdgcn_wmma_f32_16x16x4_f32(false, h1a, false, u1b,
                                                       (short)0, zx, false, false);
        #pragma unroll
        for (int r = 0; r < 8; ++r) {
            const float zv = z1[r];
            const float sg = fast_sigmoid(zv);
            const float a  = sg + act_sel * (fmaxf(zv, 0.f) - sg);
            const float gb = quad_bcast<0xAA>(a);     // g -> all quad lanes
            const float ig = a * gb;                  // lane0: i*g
            const float fb = quad_bcast<0x55>(a);     // f -> all quad lanes
            const float cn = fb * c1[r] + ig;         // lane0 valid
            c1[r] = cn;
            const float ob = quad_bcast<0xFF>(a);     // o -> all quad lanes
            hbuf[r] = ob * fmaxf(cn, 0.f);            // lane0 valid
        }
        if ((lane & 3) == 0) {
            #pragma unroll
            for (int r = 0; r < 8; ++r)
                lds1[wv][((r + (hi << 3)) << 2) | u] = hbuf[r];
        }

        // ======== off-path: x(t+1)@W1 while layer-1 stores drain ========
        {
            int tn = t + 1; if (tn >= T_SZ) tn = T_SZ - 1;   // clamp (uniform)
            const float* pn = rowbase + tn * F_SZ;
            #pragma unroll
            for (int j = 0; j < 4; ++j)
                xa[j] = *(const v2f*)(pn + 4 * j + 2 * hi);
            if (t + 4 < T_SZ) __builtin_prefetch(pn + 3 * F_SZ, 0, 3);
            zx = cbias1;
            #pragma unroll
            for (int j = 0; j < 4; ++j)
                zx = __builtin_amdgcn_wmma_f32_16x16x4_f32(false, xa[j], false, w1b[j],
                                                           (short)0, zx, false, false);
        }

        // finish layer-1 transpose
        asm volatile("s_wait_dscnt 0" ::: "memory");
        h1a.x = lds1[wv][(m << 2) + 2 * hi + 0];
        h1a.y = lds1[wv][(m << 2) + 2 * hi + 1];
        asm volatile("" ::: "memory");

        // ======== layer 2 critical segment: z2 = zh2 + h1@W2 ========
        v8f z2 = __builtin_amdgcn_wmma_f32_16x16x4_f32(false, h1a, false, w2b,
                                                       (short)0, zh2, false, false);
        #pragma unroll
        for (int r = 0; r < 8; ++r) {
            const float zv = z2[r];
            const float sg = fast_sigmoid(zv);
            const float a  = sg + act_sel * (fmaxf(zv, 0.f) - sg);
            const float gb = quad_bcast<0xAA>(a);
            const float ig = a * gb;
            const float fb = quad_bcast<0x55>(a);
            const float cn = fb * c2[r] + ig;
            c2[r] = cn;
            const float ob = quad_bcast<0xFF>(a);
            hbuf[r] = ob * fmaxf(cn, 0.f);
        }
        if ((lane & 3) == 0) {
            #pragma unroll
            for (int r = 0; r < 8; ++r)
                lds2[wv][((r + (hi << 3)) << 2) | u] = hbuf[r];
        }
        asm volatile("s_wait_dscnt 0" ::: "memory");
        h2a.x = lds2[wv][(m << 2) + 2 * hi + 0];
        h2a.y = lds2[wv][(m << 2) + 2 * hi + 1];
        asm volatile("" ::: "memory");

        // off-path for next step: zh2 = b2 + h2@U2 (overlaps next layer-1 work)
        zh2 = __builtin_amdgcn_wmma_f32_16x16x4_f32(false, h2a, false, u2b,
                                                    (short)0, cbias2, false, false);
    }

    // ---------------- FC (4 -> 1) + sigmoid ----------------
    float s = h2a.x * Wfc[2 * hi + 0] + h2a.y * Wfc[2 * hi + 1];
    s += __shfl_xor(s, 16, 32);
    if (hi == 0)
        out[b0 + m] = fast_sigmoid(s + bfc[0]);
}

extern "C" void kernel_launch(void* const* d_in, const int* in_sizes, int n_in,
                              void* d_out, int out_size, void* d_ws, size_t ws_size,
                              hipStream_t stream) {
    (void)in_sizes; (void)n_in; (void)out_size; (void)d_ws; (void)ws_size;
    const float* x   = (const float*)d_in[0];
    const float* W1  = (const float*)d_in[1];
    const float* U1  = (const float*)d_in[2];
    const float* b1  = (const float*)d_in[3];
    const float* W2  = (const float*)d_in[4];
    const float* U2  = (const float*)d_in[5];
    const float* b2  = (const float*)d_in[6];
    const float* Wfc = (const float*)d_in[7];
    const float* bfc = (const float*)d_in[8];
    float* out = (float*)d_out;

    const int waves  = B_SZ / 16;                       // 256 independent waves
    const int blocks = waves / WAVES_PER_BLOCK;         // 64 blocks x 128 threads
    lstm2_fc_wmma_f32<<<blocks, 32 * WAVES_PER_BLOCK, 0, stream>>>(
        x, W1, U1, b1, W2, U2, b2, Wfc, bfc, out);
}